// NodeAttention_27470610825503
// MI455X (gfx1250) — compile-verified
//
#include <hip/hip_runtime.h>
#include <hip/hip_bf16.h>

#define HIDDEN 128
#define SLOPE 0.2f
#define LDSW_STRIDE 132   // 128 + 4 floats pad: keeps 16B alignment, spreads banks (4*n mod 64)

typedef float v2f __attribute__((ext_vector_type(2)));
typedef float v8f __attribute__((ext_vector_type(8)));

__device__ __forceinline__ float lrelu(float v) { return v > 0.f ? v : SLOPE * v; }

// Monotone float<->uint mapping so unsigned atomicMax == float max.
__device__ __forceinline__ unsigned flipf(float f) {
    unsigned u = __float_as_uint(f);
    return (u & 0x80000000u) ? ~u : (u | 0x80000000u);
}
__device__ __forceinline__ float unflipf(unsigned u) {
    return __uint_as_float((u & 0x80000000u) ? (u & 0x7FFFFFFFu) : ~u);
}

// ---------------------------------------------------------------------------
// 0) init: zero out + denom, set emax to flip(-inf) = 0x007FFFFF
// ---------------------------------------------------------------------------
__global__ void init_kernel(float* __restrict__ out, float* __restrict__ denom,
                            unsigned* __restrict__ emax_u, int out_n, int n_nodes) {
    int i = blockIdx.x * blockDim.x + threadIdx.x;
    if (i < out_n) out[i] = 0.f;
    if (i < n_nodes) {
        denom[i] = 0.f;
        emax_u[i] = 0x007FFFFFu; // flipf(-inf)
    }
}

// ---------------------------------------------------------------------------
// 1) GEMM: hl = x @ Wl^T, hr = x @ Wr^T via V_WMMA_F32_16X16X4_F32.
//    Block = 256 threads = 8 waves; block stages its W matrix (64KB) in LDS.
//    Each wave computes a full 16x128 row strip with 8 independent
//    accumulators -> 8 independent WMMAs per k-step (no C-chain stalls),
//    A loaded once per k-step (float2), B fed from LDS (ds_load_b64).
//    Layouts (per ISA 7.12.2):
//      A(16x4): lane -> (m = lane&15, kpair = (lane>>4)*2): A[m][kp], A[m][kp+1]
//      B(4x16): lane -> (n = lane&15, kpair): B[kp][n], B[kp+1][n]; B[k][n] = W[n][k]
//      C/D:     VGPR r -> M = r + (lane>>4)*8, N = lane&15
// ---------------------------------------------------------------------------
__global__ __launch_bounds__(256) void gemm_wmma_kernel(
        const float* __restrict__ x,
        const float* __restrict__ Wl,
        const float* __restrict__ Wr,
        float* __restrict__ hl,
        float* __restrict__ hr,
        int n_nodes, int blocksPerMat) {
    __shared__ float sW[HIDDEN * LDSW_STRIDE]; // 67,584 B (well under 320KB/WGP)

    const int which = blockIdx.x / blocksPerMat;
    const int mb = blockIdx.x % blocksPerMat;
    const float* __restrict__ W = which ? Wr : Wl;
    float* __restrict__ H = which ? hr : hl;

    // Stage W into LDS: 4096 float4s, 16 per thread, coalesced.
    for (int i = threadIdx.x; i < HIDDEN * (HIDDEN / 4); i += 256) {
        const int n = i >> 5;           // row (32 float4 per row)
        const int k4 = (i & 31) << 2;   // col
        const float4 v = *(const float4*)(W + (size_t)n * HIDDEN + k4);
        *(float4*)(sW + n * LDSW_STRIDE + k4) = v;
    }
    __syncthreads();

    const int mTiles = (n_nodes + 15) >> 4;
    const int mt = mb * 8 + (threadIdx.x >> 5);
    if (mt >= mTiles) return;
    const int lane = threadIdx.x & 31;
    const int lm = lane & 15;
    const int kp = (lane >> 4) << 1;

    int arow = mt * 16 + lm;
    if (arow >= n_nodes) arow = n_nodes - 1; // clamp; stores guarded on tail tile
    const float* __restrict__ aPtr = x + (size_t)arow * HIDDEN + kp;
    const float* __restrict__ bPtr = sW + lm * LDSW_STRIDE + kp;

    v8f acc[8] = {};
#pragma unroll 4
    for (int k = 0; k < HIDDEN; k += 4) {
        const v2f a = *(const v2f*)(aPtr + k);
#pragma unroll
        for (int nt = 0; nt < 8; ++nt) {
            const v2f b = *(const v2f*)(bPtr + nt * 16 * LDSW_STRIDE + k);
            acc[nt] = __builtin_amdgcn_wmma_f32_16x16x4_f32(
                false, a, false, b, (short)0, acc[nt], false, false);
        }
    }

    // Store 16x128 strip. Wave-uniform tail check -> branchless fast path.
    const int mbase = mt * 16 + ((lane >> 4) << 3);
    float* __restrict__ oPtr = H + (size_t)mbase * HIDDEN + lm;
    if (mt * 16 + 16 <= n_nodes) {
#pragma unroll
        for (int nt = 0; nt < 8; ++nt)
#pragma unroll
            for (int r = 0; r < 8; ++r)
                oPtr[(size_t)r * HIDDEN + nt * 16] = acc[nt][r];
    } else {
#pragma unroll
        for (int nt = 0; nt < 8; ++nt)
#pragma unroll
            for (int r = 0; r < 8; ++r)
                if (mbase + r < n_nodes)
                    oPtr[(size_t)r * HIDDEN + nt * 16] = acc[nt][r];
    }
}

// ---------------------------------------------------------------------------
// 2) per-edge score: e = sum_d lrelu((hl[src][d] + hr[dst][d]) * alpha)
//    One wave per edge; lane handles 4 contiguous floats (b128 load).
// ---------------------------------------------------------------------------
__global__ void edge_score_kernel(const float* __restrict__ hl,
                                  const float* __restrict__ hr,
                                  const float* __restrict__ alpha,
                                  const int* __restrict__ src,
                                  const int* __restrict__ dst,
                                  float* __restrict__ e_out,
                                  unsigned* __restrict__ emax_u,
                                  int n_edges) {
    const int wave = blockIdx.x * (blockDim.x >> 5) + (threadIdx.x >> 5);
    if (wave >= n_edges) return;
    const int lane = threadIdx.x & 31;

    const int s = src[wave];
    const int d = dst[wave];
    const float al = alpha[wave];

    const float4 a = *(const float4*)(hl + (size_t)s * HIDDEN + lane * 4);
    const float4 b = *(const float4*)(hr + (size_t)d * HIDDEN + lane * 4);

    float sum = lrelu((a.x + b.x) * al) + lrelu((a.y + b.y) * al) +
                lrelu((a.z + b.z) * al) + lrelu((a.w + b.w) * al);

#pragma unroll
    for (int off = 16; off > 0; off >>= 1) sum += __shfl_down(sum, off, 32);

    if (lane == 0) {
        e_out[wave] = sum;
        atomicMax(emax_u + d, flipf(sum));
    }
}

// ---------------------------------------------------------------------------
// 3) ex = exp(e - emax[dst]); denom[dst] += ex. One thread per edge.
// ---------------------------------------------------------------------------
__global__ void edge_exp_kernel(const float* __restrict__ e_in,
                                const int* __restrict__ dst,
                                const unsigned* __restrict__ emax_u,
                                float* __restrict__ ex_out,
                                float* __restrict__ denom,
                                int n_edges) {
    const int i = blockIdx.x * blockDim.x + threadIdx.x;
    if (i >= n_edges) return;
    const int d = dst[i];
    const float ex = __expf(e_in[i] - unflipf(emax_u[d]));
    ex_out[i] = ex;
    atomicAdd(denom + d, ex);
}

// ---------------------------------------------------------------------------
// 4) scatter: out[dst] += x[src] * (ex / denom[dst]). One wave per edge,
//    float4 gather, 4 scalar float atomics per lane (global_atomic_add_f32).
// ---------------------------------------------------------------------------
__global__ void edge_scatter_kernel(const float* __restrict__ x,
                                    const int* __restrict__ src,
                                    const int* __restrict__ dst,
                                    const float* __restrict__ ex_in,
                                    const float* __restrict__ denom,
                                    float* __restrict__ out,
                                    int n_edges) {
    const int wave = blockIdx.x * (blockDim.x >> 5) + (threadIdx.x >> 5);
    if (wave >= n_edges) return;
    const int lane = threadIdx.x & 31;

    const int s = src[wave];
    const int d = dst[wave];
    const float a = ex_in[wave] / denom[d];

    const float4 xv = *(const float4*)(x + (size_t)s * HIDDEN + lane * 4);
    float* o = out + (size_t)d * HIDDEN + lane * 4;
    atomicAdd(o + 0, xv.x * a);
    atomicAdd(o + 1, xv.y * a);
    atomicAdd(o + 2, xv.z * a);
    atomicAdd(o + 3, xv.w * a);
}

// ---------------------------------------------------------------------------
// launch
// ---------------------------------------------------------------------------
extern "C" void kernel_launch(void* const* d_in, const int* in_sizes, int n_in,
                              void* d_out, int out_size, void* d_ws, size_t ws_size,
                              hipStream_t stream) {
    const float* x     = (const float*)d_in[0];
    const float* W_l   = (const float*)d_in[1];
    const float* W_r   = (const float*)d_in[2];
    const float* alpha = (const float*)d_in[3];
    const int*   src   = (const int*)d_in[4];
    const int*   dst   = (const int*)d_in[5];
    float* out = (float*)d_out;

    const int n_nodes = in_sizes[0] / HIDDEN;
    const int n_edges = in_sizes[4];

    // Workspace layout (floats): hl[N*H] hr[N*H] e[E] ex[E] denom[N] emax[N]
    float* hl = (float*)d_ws;
    float* hr = hl + (size_t)n_nodes * HIDDEN;
    float* e_buf = hr + (size_t)n_nodes * HIDDEN;
    float* ex_buf = e_buf + n_edges;
    float* denom = ex_buf + n_edges;
    unsigned* emax_u = (unsigned*)(denom + n_nodes);

    const int out_n = n_nodes * HIDDEN;

    // 0) init
    {
        int threads = 256;
        int blocks = (out_n + threads - 1) / threads;
        init_kernel<<<blocks, threads, 0, stream>>>(out, denom, emax_u, out_n, n_nodes);
    }
    // 1) GEMMs (both matrices in one launch; 8 row-strips of 16 per block)
    {
        const int mTiles = (n_nodes + 15) / 16;
        const int blocksPerMat = (mTiles + 7) / 8;
        gemm_wmma_kernel<<<blocksPerMat * 2, 256, 0, stream>>>(
            x, W_l, W_r, hl, hr, n_nodes, blocksPerMat);
    }
    // 2) per-edge scores + segment max
    {
        const int wavesPerBlock = 8;
        int blocks = (n_edges + wavesPerBlock - 1) / wavesPerBlock;
        edge_score_kernel<<<blocks, wavesPerBlock * 32, 0, stream>>>(
            hl, hr, alpha, src, dst, e_buf, emax_u, n_edges);
    }
    // 3) exp + denom
    {
        int threads = 256;
        int blocks = (n_edges + threads - 1) / threads;
        edge_exp_kernel<<<blocks, threads, 0, stream>>>(e_buf, dst, emax_u, ex_buf, denom, n_edges);
    }
    // 4) weighted scatter
    {
        const int wavesPerBlock = 8;
        int blocks = (n_edges + wavesPerBlock - 1) / wavesPerBlock;
        edge_scatter_kernel<<<blocks, wavesPerBlock * 32, 0, stream>>>(
            x, src, dst, ex_buf, denom, out, n_edges);
    }
}